// GCNNet_20117626814681
// MI455X (gfx1250) — compile-verified
//
#include <hip/hip_runtime.h>
#include <hip/hip_bf16.h>

typedef __attribute__((ext_vector_type(2))) float v2f;
typedef __attribute__((ext_vector_type(8))) float v8f;

#define IN_CH   128
#define HID_CH  256
#define OUT_CH  16

// ---------------------------------------------------------------------------
// Utility: zero a float buffer (float4 grid-stride)
// ---------------------------------------------------------------------------
__global__ void zero_f4(float4* p, size_t n4) {
    size_t i = (size_t)blockIdx.x * blockDim.x + threadIdx.x;
    size_t stride = (size_t)gridDim.x * blockDim.x;
    float4 z = make_float4(0.f, 0.f, 0.f, 0.f);
    for (; i < n4; i += stride) p[i] = z;
}

// ---------------------------------------------------------------------------
// Degree accumulation: deg[col[e]] += 1
// ---------------------------------------------------------------------------
__global__ void deg_kernel(const int* __restrict__ col, float* __restrict__ deg, int E) {
    int e = blockIdx.x * blockDim.x + threadIdx.x;
    if (e < E) unsafeAtomicAdd(deg + col[e], 1.0f);
}

// deg -> dinv in place: d > 0 ? rsqrt(d) : 0
__global__ void dinv_kernel(float* __restrict__ d, int N) {
    int i = blockIdx.x * blockDim.x + threadIdx.x;
    if (i < N) {
        float v = d[i];
        d[i] = (v > 0.f) ? __frsqrt_rn(v) : 0.f;
    }
}

// norm[e] = dinv[row[e]] * dinv[col[e]]
__global__ void norm_kernel(const int* __restrict__ row, const int* __restrict__ col,
                            const float* __restrict__ dinv, float* __restrict__ norm, int E) {
    int e = blockIdx.x * blockDim.x + threadIdx.x;
    if (e < E) norm[e] = dinv[row[e]] * dinv[col[e]];
}

// ---------------------------------------------------------------------------
// Pack W [K,F] row-major into WMMA B-fragment order:
//   float2 slot (nt, kk4, lane): lane n<16 -> K = 4*kk4 + {0,1}, N = 16*nt + n
//                                lane n>=16 -> K = 4*kk4 + {2,3}, N = 16*nt + n-16
// ---------------------------------------------------------------------------
__global__ void pack_w(const float* __restrict__ W, float* __restrict__ Wp, int K, int F) {
    int idx = blockIdx.x * blockDim.x + threadIdx.x;   // over (F/16)*(K/4)*32
    int total = (F / 16) * (K / 4) * 32;
    if (idx >= total) return;
    int lane = idx & 31;
    int kk4  = (idx >> 5) % (K / 4);
    int nt   = (idx >> 5) / (K / 4);
    int n = nt * 16 + (lane & 15);
    int k = kk4 * 4 + 2 * (lane >> 4);
    Wp[2 * idx + 0] = W[(size_t)k * F + n];
    Wp[2 * idx + 1] = W[(size_t)(k + 1) * F + n];
}

// ---------------------------------------------------------------------------
// WMMA fp32 GEMM:  C[N,F] = H[N,K] @ W[K,F]
//   grid.x = N/16 M-tiles; block = 256 (8 waves)
//   block stages its 16xK A-tile in LDS (stride K+4 -> conflict-free b64 reads),
//   waves cover N-tiles with v_wmma_f32_16x16x4_f32 accumulate over K.
// ---------------------------------------------------------------------------
__global__ __launch_bounds__(256) void gemm_wmma_f32(
        const float* __restrict__ H, const float* __restrict__ Wp,
        float* __restrict__ C, int K, int NT, int F) {
    __shared__ float As[16 * (256 + 4)];            // max K = 256, padded stride
    const int tid  = threadIdx.x;
    const int m0   = blockIdx.x * 16;
    const int ldk  = K + 4;

    // cooperative coalesced stage of A tile
    const int total = 16 * K;
    for (int idx = tid; idx < total; idx += 256) {
        int r  = idx / K;
        int ck = idx - r * K;
        As[r * ldk + ck] = H[(size_t)(m0 + r) * K + ck];
    }
    __syncthreads();

    const int wave = tid >> 5;
    const int lane = tid & 31;
    const int abase = (lane & 15) * ldk + 2 * (lane >> 4);
    const int k4 = K >> 2;

    for (int nt = wave; nt < NT; nt += 8) {
        v8f acc = {};
        const v2f* bp = (const v2f*)Wp + (size_t)nt * k4 * 32 + lane;
        #pragma unroll 4
        for (int kk = 0; kk < K; kk += 4) {
            v2f a = *(const v2f*)(As + abase + kk);   // ds_load_b64
            v2f b = *bp;                              // global_load_b64, coalesced
            bp += 32;
            acc = __builtin_amdgcn_wmma_f32_16x16x4_f32(
                false, a, false, b, (short)0, acc, false, false);
        }
        // D layout: VGPR j, lane n  ->  row m0 + j + 8*(n>>4), col nt*16 + (n&15)
        const int mrow = m0 + ((lane >> 4) << 3);
        const int ncol = nt * 16 + (lane & 15);
        #pragma unroll
        for (int j = 0; j < 8; j++)
            C[(size_t)(mrow + j) * F + ncol] = acc[j];
    }
}

// ---------------------------------------------------------------------------
// Edge scatter, F=256: one wave per edge; 8 channels per lane.
//   out[col[e], :] += norm[e] * t[row[e], :]
// ---------------------------------------------------------------------------
__global__ __launch_bounds__(256) void scatter256(
        const float* __restrict__ t, const int* __restrict__ row,
        const int* __restrict__ col, const float* __restrict__ norm,
        float* __restrict__ out, int E) {
    int gid  = blockIdx.x * blockDim.x + threadIdx.x;
    int e    = gid >> 5;
    int lane = gid & 31;
    if (e >= E) return;
    int r = row[e], c = col[e];
    float w = norm[e];
    const float4* src = (const float4*)(t + (size_t)r * 256) + lane * 2;
    float4 v0 = src[0];
    float4 v1 = src[1];
    float* dst = out + (size_t)c * 256 + lane * 8;
    unsafeAtomicAdd(dst + 0, w * v0.x);
    unsafeAtomicAdd(dst + 1, w * v0.y);
    unsafeAtomicAdd(dst + 2, w * v0.z);
    unsafeAtomicAdd(dst + 3, w * v0.w);
    unsafeAtomicAdd(dst + 4, w * v1.x);
    unsafeAtomicAdd(dst + 5, w * v1.y);
    unsafeAtomicAdd(dst + 6, w * v1.z);
    unsafeAtomicAdd(dst + 7, w * v1.w);
}

// Edge scatter, F=16: one thread per edge.
__global__ void scatter16(
        const float* __restrict__ t, const int* __restrict__ row,
        const int* __restrict__ col, const float* __restrict__ norm,
        float* __restrict__ out, int E) {
    int e = blockIdx.x * blockDim.x + threadIdx.x;
    if (e >= E) return;
    int r = row[e], c = col[e];
    float w = norm[e];
    const float4* src = (const float4*)(t + (size_t)r * 16);
    float* dst = out + (size_t)c * 16;
    #pragma unroll
    for (int q = 0; q < 4; q++) {
        float4 v = src[q];
        unsafeAtomicAdd(dst + 4 * q + 0, w * v.x);
        unsafeAtomicAdd(dst + 4 * q + 1, w * v.y);
        unsafeAtomicAdd(dst + 4 * q + 2, w * v.z);
        unsafeAtomicAdd(dst + 4 * q + 3, w * v.w);
    }
}

// out[i] = leaky_relu(out[i] + b[i % 256]), slope 0.01
__global__ void bias_lrelu256(float* __restrict__ buf, const float* __restrict__ b, size_t total) {
    size_t i = (size_t)blockIdx.x * blockDim.x + threadIdx.x;
    size_t stride = (size_t)gridDim.x * blockDim.x;
    for (; i < total; i += stride) {
        float v = buf[i] + b[i & 255];
        buf[i] = (v > 0.f) ? v : 0.01f * v;
    }
}

// Final: zero nodes (batch-change positions) -> output row = batch id, + b3
__global__ void gather_out(const float* __restrict__ h, const int* __restrict__ batch,
                           const float* __restrict__ b3, float* __restrict__ out, int N) {
    int i = blockIdx.x * blockDim.x + threadIdx.x;
    if (i >= N) return;
    bool zero = (i == 0) || (batch[i] != batch[i - 1]);
    if (!zero) return;
    int g = batch[i];
    #pragma unroll
    for (int j = 0; j < 16; j++)
        out[(size_t)g * 16 + j] = h[(size_t)i * 16 + j] + b3[j];
}

// ---------------------------------------------------------------------------
extern "C" void kernel_launch(void* const* d_in, const int* in_sizes, int n_in,
                              void* d_out, int out_size, void* d_ws, size_t ws_size,
                              hipStream_t stream) {
    const float* x    = (const float*)d_in[0];
    const int*   eidx = (const int*)d_in[1];
    const int*   batch= (const int*)d_in[2];
    const float* W1   = (const float*)d_in[3];
    const float* b1   = (const float*)d_in[4];
    const float* W2   = (const float*)d_in[5];
    const float* b2   = (const float*)d_in[6];
    const float* W3   = (const float*)d_in[7];
    const float* b3   = (const float*)d_in[8];

    const int E = in_sizes[1] / 2;
    const int N = in_sizes[0] / IN_CH;
    const int* row = eidx;
    const int* col = eidx + E;

    // workspace carve-up (256-byte aligned)
    char* ws = (char*)d_ws;
    size_t off = 0;
    auto carve = [&](size_t bytes) {
        char* p = ws + off;
        off += (bytes + 255) & ~(size_t)255;
        return p;
    };
    float* dinv  = (float*)carve((size_t)N * 4);
    float* norm  = (float*)carve((size_t)E * 4);
    float* Wp1   = (float*)carve((size_t)IN_CH  * HID_CH * 4);
    float* Wp2   = (float*)carve((size_t)HID_CH * HID_CH * 4);
    float* Wp3   = (float*)carve((size_t)HID_CH * OUT_CH * 4);
    float* bufA  = (float*)carve((size_t)N * HID_CH * 4);
    float* bufB  = (float*)carve((size_t)N * HID_CH * 4);
    float* buf16a= (float*)carve((size_t)N * OUT_CH * 4);
    float* buf16b= (float*)carve((size_t)N * OUT_CH * 4);
    (void)ws_size;

    const int eb = (E + 255) / 256;
    const int nb = (N + 255) / 256;
    const int mtiles = N / 16;                     // 3125
    const size_t wideN4 = (size_t)N * HID_CH / 4;  // float4 count
    const size_t out16N4 = (size_t)N * OUT_CH / 4;

    // --- normalization coefficients ---
    hipLaunchKernelGGL(zero_f4, dim3((unsigned)((N / 4 + 255) / 256)), dim3(256), 0, stream,
                       (float4*)dinv, (size_t)N / 4);
    hipLaunchKernelGGL(deg_kernel, dim3(eb), dim3(256), 0, stream, col, dinv, E);
    hipLaunchKernelGGL(dinv_kernel, dim3(nb), dim3(256), 0, stream, dinv, N);
    hipLaunchKernelGGL(norm_kernel, dim3(eb), dim3(256), 0, stream, row, col, dinv, norm, E);

    // --- pack weights into WMMA B-fragment layout ---
    {
        int t1 = (HID_CH / 16) * (IN_CH / 4) * 32;
        int t2 = (HID_CH / 16) * (HID_CH / 4) * 32;
        int t3 = (OUT_CH / 16) * (HID_CH / 4) * 32;
        hipLaunchKernelGGL(pack_w, dim3((t1 + 255) / 256), dim3(256), 0, stream, W1, Wp1, IN_CH, HID_CH);
        hipLaunchKernelGGL(pack_w, dim3((t2 + 255) / 256), dim3(256), 0, stream, W2, Wp2, HID_CH, HID_CH);
        hipLaunchKernelGGL(pack_w, dim3((t3 + 255) / 256), dim3(256), 0, stream, W3, Wp3, HID_CH, OUT_CH);
    }

    // --- layer 1: t0 = x @ W1 ; h1 = lrelu(scatter(t0) + b1) ---
    hipLaunchKernelGGL(gemm_wmma_f32, dim3(mtiles), dim3(256), 0, stream,
                       x, Wp1, bufA, IN_CH, HID_CH / 16, HID_CH);
    hipLaunchKernelGGL(zero_f4, dim3(4096), dim3(256), 0, stream, (float4*)bufB, wideN4);
    hipLaunchKernelGGL(scatter256, dim3((E * 32 + 255) / 256), dim3(256), 0, stream,
                       bufA, row, col, norm, bufB, E);
    hipLaunchKernelGGL(bias_lrelu256, dim3(4096), dim3(256), 0, stream,
                       bufB, b1, (size_t)N * HID_CH);

    // --- layer 2: t1 = h1 @ W2 ; h2 = lrelu(scatter(t1) + b2) ---
    hipLaunchKernelGGL(gemm_wmma_f32, dim3(mtiles), dim3(256), 0, stream,
                       bufB, Wp2, bufA, HID_CH, HID_CH / 16, HID_CH);
    hipLaunchKernelGGL(zero_f4, dim3(4096), dim3(256), 0, stream, (float4*)bufB, wideN4);
    hipLaunchKernelGGL(scatter256, dim3((E * 32 + 255) / 256), dim3(256), 0, stream,
                       bufA, row, col, norm, bufB, E);
    hipLaunchKernelGGL(bias_lrelu256, dim3(4096), dim3(256), 0, stream,
                       bufB, b2, (size_t)N * HID_CH);

    // --- layer 3: t2 = h2 @ W3 ; h3 = scatter(t2) (+ b3 folded into gather) ---
    hipLaunchKernelGGL(gemm_wmma_f32, dim3(mtiles), dim3(256), 0, stream,
                       bufB, Wp3, buf16a, HID_CH, OUT_CH / 16, OUT_CH);
    hipLaunchKernelGGL(zero_f4, dim3(1024), dim3(256), 0, stream, (float4*)buf16b, out16N4);
    hipLaunchKernelGGL(scatter16, dim3(eb), dim3(256), 0, stream,
                       buf16a, row, col, norm, buf16b, E);

    // --- readout: first node of each graph, + b3 ---
    hipLaunchKernelGGL(gather_out, dim3(nb), dim3(256), 0, stream,
                       buf16b, batch, b3, (float*)d_out, N);
}